// _FastFoodProjection_635655159910
// MI455X (gfx1250) — compile-verified
//
#include <hip/hip_runtime.h>

// FastFood projection for MI455X (gfx1250, wave32).
// Two length-2^25 FWHTs, each split into:
//   pass A: 15 low bits, contiguous 32768-float blocks in 128KB LDS (all WMMA)
//   pass B: 10 high bits, 1024-row x 32-col strided tiles in 128KB LDS (all WMMA)
// H16 applied as chained V_WMMA_F32_16X16X4_F32; partial rounds use H8(x)I2 / H4(x)I4.
// All LDS accesses are written directly against the __shared__ symbol (macros, not
// function parameters) so the compiler emits 32-bit addrspace(3) addressing.
// Scratch: needs 2 * 2^25 * 4B = 256 MB in d_ws.

#define SIZE_LOG   25
#define FFSIZE     (1u << SIZE_LOG)        // 33554432
#define FLAT_DIM   30000000u
#define FLAT4      (FLAT_DIM / 4u)         // exact: 7.5M
#define PASSA_LOG  15
#define PASSA_N    (1 << PASSA_LOG)        // 32768 floats = 128 KB LDS
#define NCOLS      32                      // pass-B columns per workgroup
#define THREADS    512
#define WAVES      (THREADS / 32)
#define TILES_PER_ROUND (PASSA_N / 256)    // 128 tiles of 16x16 per round
#define TILES_PER_WAVE  (TILES_PER_ROUND / WAVES)  // 8

typedef __attribute__((ext_vector_type(2))) float v2f;
typedef __attribute__((ext_vector_type(4))) float v4f;
typedef __attribute__((ext_vector_type(8))) float v8f;
typedef __attribute__((ext_vector_type(4))) int   v4i;

// Build the A-matrix fragment for H_{16>>s} (x) I_{2^s}.
// A is 16x4 per WMMA chunk; ISA layout: VGPR0 = K(0|2), VGPR1 = K(1|3) across lane halves.
__device__ inline void make_A(v2f A[4], int lane, int s) {
    const int m    = lane & 15;
    const int koff = (lane >> 4) & 1;
    const int mask = (1 << s) - 1;
#pragma unroll
    for (int c = 0; c < 4; ++c) {
        const int k0 = 4 * c + 2 * koff;
#pragma unroll
        for (int j = 0; j < 2; ++j) {
            const int k = k0 + j;
            float v = 0.0f;
            if ((m & mask) == (k & mask))
                v = (__builtin_popcount((m >> s) & (k >> s)) & 1) ? -1.0f : 1.0f;
            A[c][j] = v;
        }
    }
}

// One WMMA round over 128 disjoint 16x16 tiles of `lds`.
// ADDR is an expression in (t, r, c) giving the LDS float index.
// Requires in scope: lds (the __shared__ array), wave, koff, m.
#define H16_ROUND(AF, ADDR)                                                    \
  do {                                                                         \
    _Pragma("unroll")                                                          \
    for (int _i = 0; _i < TILES_PER_WAVE; ++_i) {                              \
      const int t = wave * TILES_PER_WAVE + _i;                                \
      v2f _Bv[4];                                                              \
      _Pragma("unroll")                                                        \
      for (int _c = 0; _c < 4; ++_c) {                                         \
        {                                                                      \
          const int r = 4 * _c + 2 * koff;                                     \
          const int c = m;                                                     \
          _Bv[_c][0] = lds[(ADDR)];                                            \
        }                                                                      \
        {                                                                      \
          const int r = 4 * _c + 2 * koff + 1;                                 \
          const int c = m;                                                     \
          _Bv[_c][1] = lds[(ADDR)];                                            \
        }                                                                      \
      }                                                                        \
      v8f _acc = {0.f, 0.f, 0.f, 0.f, 0.f, 0.f, 0.f, 0.f};                     \
      _Pragma("unroll")                                                        \
      for (int _c = 0; _c < 4; ++_c)                                           \
        _acc = __builtin_amdgcn_wmma_f32_16x16x4_f32(                          \
            false, AF[_c], false, _Bv[_c], (short)0, _acc, false, false);      \
      _Pragma("unroll")                                                        \
      for (int _v = 0; _v < 8; ++_v) {                                         \
        const int r = _v + 8 * koff;                                           \
        const int c = m;                                                       \
        lds[(ADDR)] = _acc[_v];                                                \
      }                                                                        \
    }                                                                          \
  } while (0)

// 15-stage FWHT on a linear 32768-float LDS block (index bits 0..14).
#define FWHT15_BODY()                                                          \
  do {                                                                         \
    __syncthreads();                                                           \
    H16_ROUND(A16, (t << 8) | (c << 4) | r);                   /* bits 0-3  */ \
    __syncthreads();                                                           \
    H16_ROUND(A16, (t << 8) | (r << 4) | c);                   /* bits 4-7  */ \
    __syncthreads();                                                           \
    H16_ROUND(A16, ((t >> 4) << 12) | (r << 8) | ((t & 15) << 4) | c);         \
    __syncthreads();                                           /* bits 8-11 */ \
    H16_ROUND(A8, (r << 11) | (t << 4) | c);    /* bits 12-14 via H8(x)I2  */  \
    __syncthreads();                                                           \
  } while (0)

// 10-stage FWHT over the row dim of a 1024-row x 32-col LDS tile (idx = r*32+c).
#define FWHT10_BODY()                                                          \
  do {                                                                         \
    __syncthreads();                                                           \
    H16_ROUND(A16, ((((t >> 1) << 4) | r) << 5) | ((t & 1) << 4) | c);         \
    __syncthreads();                                          /* r bits 0-3 */ \
    H16_ROUND(A16, ((((t >> 5) << 8) | (r << 4) | ((t >> 1) & 15)) << 5) |     \
                       ((t & 1) << 4) | c);                   /* r bits 4-7 */ \
    __syncthreads();                                                           \
    H16_ROUND(A4, (((r << 6) | (t >> 1)) << 5) | ((t & 1) << 4) | c);          \
    __syncthreads();                            /* r bits 8-9 via H4(x)I4  */  \
  } while (0)

// Kernel 1: x1 = FWHT_low15( pad(theta) * B ), contiguous blocks, float4 I/O.
__global__ __launch_bounds__(THREADS)
void k_fwht_a_first(const float* __restrict__ theta, const float* __restrict__ Bsign,
                    float* __restrict__ out) {
    __shared__ __align__(16) float lds[PASSA_N];
    const int tid = threadIdx.x, lane = tid & 31, wave = tid >> 5;
    const int m = lane & 15, koff = (lane >> 4) & 1;
    v2f A16[4], A8[4];
    make_A(A16, lane, 0);
    make_A(A8,  lane, 1);

    const unsigned base4 = ((unsigned)blockIdx.x << PASSA_LOG) >> 2;
    const v4f* T4 = (const v4f*)theta;
    const v4f* B4 = (const v4f*)Bsign;
    const v4f zero = {0.f, 0.f, 0.f, 0.f};
    for (int i = tid; i < PASSA_N / 4; i += THREADS) {
        const unsigned g4 = base4 + (unsigned)i;
        const v4f tv = (g4 < FLAT4) ? T4[g4] : zero;
        ((v4f*)lds)[i] = tv * B4[g4];
    }
    FWHT15_BODY();
    for (int i = tid; i < PASSA_N / 4; i += THREADS)
        ((v4f*)out)[base4 + (unsigned)i] = ((v4f*)lds)[i];
}

// Kernel 2: in-place FWHT over high 10 bits (strided columns, disjoint per block).
__global__ __launch_bounds__(THREADS)
void k_fwht_b_mid(float* __restrict__ x) {
    __shared__ __align__(16) float lds[PASSA_N];
    const int tid = threadIdx.x, lane = tid & 31, wave = tid >> 5;
    const int m = lane & 15, koff = (lane >> 4) & 1;
    v2f A16[4], A4[4];
    make_A(A16, lane, 0);
    make_A(A4,  lane, 2);

    const unsigned colb4 = ((unsigned)blockIdx.x * NCOLS) >> 2;  // float4 units
    v4f* X4 = (v4f*)x;
    for (int i = tid; i < PASSA_N / 4; i += THREADS) {
        const unsigned r = (unsigned)i >> 3, c4 = (unsigned)i & 7;
        ((v4f*)lds)[i] = X4[(r << (PASSA_LOG - 2)) + colb4 + c4];
    }
    FWHT10_BODY();
    for (int i = tid; i < PASSA_N / 4; i += THREADS) {
        const unsigned r = (unsigned)i >> 3, c4 = (unsigned)i & 7;
        X4[(r << (PASSA_LOG - 2)) + colb4 + c4] = ((v4f*)lds)[i];
    }
}

// Kernel 3: y = FWHT_low15( x1[Pi[.]] * G ), gather fused into the load.
__global__ __launch_bounds__(THREADS)
void k_fwht_a_perm(const float* __restrict__ xin, const int* __restrict__ Pi,
                   const float* __restrict__ G, float* __restrict__ out) {
    __shared__ __align__(16) float lds[PASSA_N];
    const int tid = threadIdx.x, lane = tid & 31, wave = tid >> 5;
    const int m = lane & 15, koff = (lane >> 4) & 1;
    v2f A16[4], A8[4];
    make_A(A16, lane, 0);
    make_A(A8,  lane, 1);

    const unsigned base4 = ((unsigned)blockIdx.x << PASSA_LOG) >> 2;
    const v4i* P4 = (const v4i*)Pi;
    const v4f* G4 = (const v4f*)G;
    for (int i = tid; i < PASSA_N / 4; i += THREADS) {
        const unsigned g4 = base4 + (unsigned)i;
        const v4i p  = P4[g4];
        const v4f gg = G4[g4];
        v4f y;
#pragma unroll
        for (int j = 0; j < 4; ++j)
            y[j] = xin[(unsigned)p[j]] * gg[j];
        ((v4f*)lds)[i] = y;
    }
    FWHT15_BODY();
    for (int i = tid; i < PASSA_N / 4; i += THREADS)
        ((v4f*)out)[base4 + (unsigned)i] = ((v4f*)lds)[i];
}

// Kernel 4: final high-10-bit FWHT + truncate to FLAT_DIM + scale.
__global__ __launch_bounds__(THREADS)
void k_fwht_b_last(const float* __restrict__ xin, const float* __restrict__ divisor,
                   float* __restrict__ out) {
    __shared__ __align__(16) float lds[PASSA_N];
    const int tid = threadIdx.x, lane = tid & 31, wave = tid >> 5;
    const int m = lane & 15, koff = (lane >> 4) & 1;
    v2f A16[4], A4[4];
    make_A(A16, lane, 0);
    make_A(A4,  lane, 2);

    const unsigned colb4 = ((unsigned)blockIdx.x * NCOLS) >> 2;
    const v4f* X4 = (const v4f*)xin;
    for (int i = tid; i < PASSA_N / 4; i += THREADS) {
        const unsigned r = (unsigned)i >> 3, c4 = (unsigned)i & 7;
        ((v4f*)lds)[i] = X4[(r << (PASSA_LOG - 2)) + colb4 + c4];
    }
    FWHT10_BODY();
    const float inv_scale =
        1.0f / (divisor[0] * __builtin_sqrtf((float)FLAT_DIM / (float)FFSIZE));
    for (int i = tid; i < PASSA_N / 4; i += THREADS) {
        const unsigned r = (unsigned)i >> 3, c4 = (unsigned)i & 7;
        const unsigned g4 = (r << (PASSA_LOG - 2)) + colb4 + c4;
        if (g4 < FLAT4) ((v4f*)out)[g4] = ((v4f*)lds)[i] * inv_scale;
    }
}

extern "C" void kernel_launch(void* const* d_in, const int* in_sizes, int n_in,
                              void* d_out, int out_size, void* d_ws, size_t ws_size,
                              hipStream_t stream) {
    const float* theta   = (const float*)d_in[0];
    const float* G       = (const float*)d_in[1];
    const float* Bsign   = (const float*)d_in[2];
    const float* divisor = (const float*)d_in[3];
    const int*   Pi      = (const int*)d_in[4];
    float* out  = (float*)d_out;
    float* buf1 = (float*)d_ws;            // 2^25 floats
    float* buf2 = buf1 + FFSIZE;           // 2^25 floats

    const dim3 grid(FFSIZE / PASSA_N);     // 1024 workgroups
    const dim3 block(THREADS);

    k_fwht_a_first<<<grid, block, 0, stream>>>(theta, Bsign, buf1);
    k_fwht_b_mid  <<<grid, block, 0, stream>>>(buf1);
    k_fwht_a_perm <<<grid, block, 0, stream>>>(buf1, Pi, G, buf2);
    k_fwht_b_last <<<grid, block, 0, stream>>>(buf2, divisor, out);
}